// NeuralExecutor2_17420387352968
// MI455X (gfx1250) — compile-verified
//
#include <hip/hip_runtime.h>

#define NB   4
#define NN   512
#define LAT  32

typedef __attribute__((ext_vector_type(2))) float v2f;
typedef __attribute__((ext_vector_type(8))) float v8f;

__device__ __forceinline__ v8f wmma4(v2f a, v2f b, v8f c) {
  // V_WMMA_F32_16X16X4_F32: D = A(16x4) * B(4x16) + C(16x16), fp32 throughout.
  return __builtin_amdgcn_wmma_f32_16x16x4_f32(false, a, false, b, (short)0, c,
                                               false, false);
}

// ---------------------------------------------------------------------------
// Precompute folded small vectors:
//  vm  = We_row @ Wm[64:96]   (32)    vtm = We_row @ Wtm[64:96] (32)
//  u1  = Wtu[0:32]  @ Wt      (32)    u2  = Wtu[32:64] @ Wt     (32)
//  vp  = We_row @ Wp[64:96]   (1)
// sm layout: [vm(32) | vtm(32) | u1(32) | u2(32) | vp(1)]
// ---------------------------------------------------------------------------
__global__ void __launch_bounds__(32) pre_kernel(
    const float* __restrict__ We, const float* __restrict__ Wm,
    const float* __restrict__ Wtm, const float* __restrict__ Wp,
    const float* __restrict__ Wtu, const float* __restrict__ Wt,
    float* __restrict__ sm) {
  int k = threadIdx.x;  // 0..31
  float a = 0.f, b = 0.f, u1 = 0.f, u2 = 0.f;
  for (int c = 0; c < 32; ++c) {
    float we = We[c];
    a  += we * Wm[(64 + c) * 32 + k];
    b  += we * Wtm[(64 + c) * 32 + k];
    u1 += Wtu[k * 32 + c] * Wt[c];
    u2 += Wtu[(32 + k) * 32 + c] * Wt[c];
  }
  sm[k] = a; sm[32 + k] = b; sm[64 + k] = u1; sm[96 + k] = u2;
  if (k == 0) {
    float vp = 0.f;
    for (int c = 0; c < 32; ++c) vp += We[c] * Wp[64 + c];
    sm[128] = vp;
  }
}

// ---------------------------------------------------------------------------
// z = concat([x(3), h(32)]) @ Wn(35,32); one wave per 16x16 tile, K padded 36.
// A layout (16x4 f32): lanes 0-15 hold K=k,k+1; lanes 16-31 hold K=k+2,k+3.
// B layout (4x16 f32): vgpr0 = rows k (lanes 0-15) / k+2 (16-31); vgpr1 = k+1/k+3.
// ---------------------------------------------------------------------------
__device__ __forceinline__ float zloadA(const float* __restrict__ x,
                                        const float* __restrict__ h,
                                        int row, int kk) {
  int kx = kk < 3 ? kk : 0;
  int kh = (kk >= 3 && kk < 35) ? (kk - 3) : 0;   // clamped (speculation-safe)
  float vx = x[row * 3 + kx];
  float vh = h[row * 32 + kh];
  return kk < 3 ? vx : (kk < 35 ? vh : 0.f);
}

__device__ __forceinline__ float zloadW(const float* __restrict__ W, int kk, int col) {
  int kw = kk < 35 ? kk : 0;                      // clamped
  float v = W[kw * 32 + col];
  return kk < 35 ? v : 0.f;
}

__global__ void __launch_bounds__(32) z_wmma(
    const float* __restrict__ x, const float* __restrict__ h,
    const float* __restrict__ Wn, float* __restrict__ z) {
  const int lane = threadIdx.x;
  const int m    = lane & 15;
  const int hi   = lane >> 4;
  const int row0 = blockIdx.x * 16;
  const int col0 = blockIdx.y * 16;
  v8f acc = {};
  for (int k = 0; k < 36; k += 4) {
    int ka = k + hi * 2;
    v2f a, b;
    a.x = zloadA(x, h, row0 + m, ka);
    a.y = zloadA(x, h, row0 + m, ka + 1);
    b.x = zloadW(Wn, ka, col0 + m);
    b.y = zloadW(Wn, ka + 1, col0 + m);
    acc = wmma4(a, b, acc);
  }
  for (int r = 0; r < 8; ++r) {
    int row = row0 + r + hi * 8;                 // C/D: vgpr r = rows r / r+8
    z[row * 32 + col0 + m] = acc[r];
  }
}

// ---------------------------------------------------------------------------
// Generic out[2048,32] = A1[2048,32]@W1[32,32] (+ A2@W2) (+ resid), via WMMA.
// ---------------------------------------------------------------------------
__global__ void __launch_bounds__(32) mm32_wmma(
    const float* __restrict__ A1, const float* __restrict__ W1,
    const float* __restrict__ A2, const float* __restrict__ W2,
    const float* __restrict__ resid, float* __restrict__ out) {
  const int lane = threadIdx.x;
  const int m    = lane & 15;
  const int hi   = lane >> 4;
  const int row0 = blockIdx.x * 16;
  const int col0 = blockIdx.y * 16;
  v8f acc = {};
  for (int k = 0; k < 32; k += 4) {
    int ka = k + hi * 2;
    v2f a, b;
    a.x = A1[(row0 + m) * 32 + ka];
    a.y = A1[(row0 + m) * 32 + ka + 1];
    b.x = W1[ka * 32 + col0 + m];
    b.y = W1[(ka + 1) * 32 + col0 + m];
    acc = wmma4(a, b, acc);
  }
  if (A2 != nullptr) {
    for (int k = 0; k < 32; k += 4) {
      int ka = k + hi * 2;
      v2f a, b;
      a.x = A2[(row0 + m) * 32 + ka];
      a.y = A2[(row0 + m) * 32 + ka + 1];
      b.x = W2[ka * 32 + col0 + m];
      b.y = W2[(ka + 1) * 32 + col0 + m];
      acc = wmma4(a, b, acc);
    }
  }
  for (int r = 0; r < 8; ++r) {
    int row = row0 + r + hi * 8;
    float v = acc[r];
    if (resid != nullptr) v += resid[row * 32 + col0 + m];
    out[row * 32 + col0 + m] = v;
  }
}

// ---------------------------------------------------------------------------
// Masked max-aggregation with LDS staging:
//   out[b,i,k] = im[b,i,k] + max_{j in adj2(i)} (jm[b,j,k] + e[b,i,j]*vm[k])
// Block = 8 waves = 8 rows i (same batch). jm[b] is staged through LDS in
// 128-row chunks (16 KB) shared by all 8 waves (8x less L2 traffic).
// Lane = feature channel k; adjacency packed with wave32 __ballot; edge
// scalar broadcast with __shfl; jm read from LDS bank-conflict-free.
// ---------------------------------------------------------------------------
__global__ void __launch_bounds__(256) agg_kernel(
    const float* __restrict__ im, const float* __restrict__ jm,
    const float* __restrict__ vm, const float* __restrict__ e,
    const int* __restrict__ adj, float* __restrict__ out) {
  __shared__ float sjm[128 * 32];                     // 16 KB
  const int lane   = threadIdx.x;                     // 0..31
  const int tid    = threadIdx.y * 32 + lane;         // 0..255
  const int i_glob = blockIdx.x * 8 + threadIdx.y;    // 0..B*N-1
  const int b = i_glob >> 9;
  const int i = i_glob & (NN - 1);
  const float vmk = vm[lane];
  const float* __restrict__ erow = e   + ((size_t)b * NN + i) * NN;
  const int*   __restrict__ arow = adj + ((size_t)b * NN + i) * NN;
  const float* __restrict__ jmB  = jm  + (size_t)b * NN * 32;
  float best = -__builtin_inff();

  for (int jc = 0; jc < NN; jc += 128) {
    __syncthreads();
    // cooperative stage: jm[b, jc..jc+127, 0..31] -> LDS (float4 = b128 ops)
    for (int v = tid * 4; v < 128 * 32; v += 256 * 4) {
      float4 d = *(const float4*)(jmB + (size_t)jc * 32 + v);
      *(float4*)(sjm + v) = d;
    }
    __syncthreads();
    if (jc + 128 < NN)                                // gfx1250 global_prefetch_b8
      __builtin_prefetch(jmB + (size_t)(jc + 128) * 32 + tid * 16, 0, 3);

    for (int sub = 0; sub < 4; ++sub) {
      const int j0 = jc + sub * 32;
      float ev = erow[j0 + lane];
      int   av = arow[j0 + lane];
      bool  nb = (av != 0) || (j0 + lane == i);       // self-loop from adj + I
      unsigned long long mask = __ballot(nb);         // wave32: bits 0..31
      const float* __restrict__ srow = sjm + (sub * 32) * 32;
      for (int jj = 0; jj < 32; ++jj) {
        float evj  = __shfl(ev, jj, 32);
        float zv   = srow[jj * 32 + lane];
        float cand = fmaf(evj, vmk, zv);
        best = ((mask >> jj) & 1ull) ? fmaxf(best, cand) : best;
      }
    }
  }
  out[(size_t)i_glob * 32 + lane] = im[(size_t)i_glob * 32 + lane] + best;
}

// new_x[r,c] = sum_k z[r,k]*Wd[k,c] + new_h[r,k]*Wd[32+k,c];  Wd is [64,3]
__global__ void newx_kernel(const float* __restrict__ z,
                            const float* __restrict__ nh,
                            const float* __restrict__ Wd,
                            float* __restrict__ out) {
  int t = blockIdx.x * blockDim.x + threadIdx.x;
  if (t >= NB * NN * 3) return;
  int r = t / 3, c = t - r * 3;
  float s = 0.f;
  for (int k = 0; k < 32; ++k)
    s += z[r * 32 + k] * Wd[k * 3 + c] + nh[r * 32 + k] * Wd[(32 + k) * 3 + c];
  out[t] = s;
}

// tau[b] = (1/N) * sum_{i,k} (new_h*u1[k] + agg2*u2[k])
__global__ void __launch_bounds__(256) tau_kernel(
    const float* __restrict__ nh, const float* __restrict__ agg2,
    const float* __restrict__ sm, float* __restrict__ tau) {
  const int b = blockIdx.x, t = threadIdx.x;
  const float* __restrict__ u1 = sm + 64;
  const float* __restrict__ u2 = sm + 96;
  float s = 0.f;
  for (int idx = t; idx < NN * 32; idx += 256) {
    int k = idx & 31;
    s += nh[(size_t)b * NN * 32 + idx] * u1[k] +
         agg2[(size_t)b * NN * 32 + idx] * u2[k];
  }
  __shared__ float red[256];
  red[t] = s; __syncthreads();
  for (int off = 128; off > 0; off >>= 1) {
    if (t < off) red[t] += red[t + off];
    __syncthreads();
  }
  if (t == 0) tau[b] = red[0] * (1.0f / NN);
}

// pi[r] = new_h[r]@Wp[0:32];  pj[r] = new_h[r]@Wp[32:64]
// One wave per row: coalesced 128B row load + wave32 butterfly reduction.
__global__ void __launch_bounds__(256) pipj_kernel(
    const float* __restrict__ nh, const float* __restrict__ Wp,
    float* __restrict__ pi, float* __restrict__ pj) {
  const int lane = threadIdx.x;                  // 0..31 = channel
  const int r    = blockIdx.x * 8 + threadIdx.y; // row
  float hv = nh[(size_t)r * 32 + lane];
  float a = hv * Wp[lane];
  float b = hv * Wp[32 + lane];
  for (int off = 16; off > 0; off >>= 1) {
    a += __shfl_xor(a, off, 32);
    b += __shfl_xor(b, off, 32);
  }
  if (lane == 0) { pi[r] = a; pj[r] = b; }
}

// p[b,i,j] = adj2 ? pi[i] + pj[j] + e*vp : -inf   (float4/int4 vectorized)
__global__ void __launch_bounds__(256) p_kernel(
    const float* __restrict__ pi, const float* __restrict__ pj,
    const float* __restrict__ e, const int* __restrict__ adj,
    const float* __restrict__ vp_s, float* __restrict__ p) {
  unsigned q   = blockIdx.x * 256u + threadIdx.x;  // over B*N*N/4 = 2^18
  unsigned idx = q * 4u;
  int j0 = idx & (NN - 1);
  int i  = (idx >> 9) & (NN - 1);
  int b  = idx >> 18;
  const float vp  = vp_s[0];
  const float pii = pi[b * NN + i];
  float4 ev = ((const float4*)e)[q];
  int4   av = ((const int4*)adj)[q];
  float4 pjv = ((const float4*)(pj + b * NN))[q & (NN / 4 - 1)];
  const float ninf = -__builtin_inff();
  float4 r;
  r.x = (av.x != 0 || i == j0    ) ? pii + pjv.x + ev.x * vp : ninf;
  r.y = (av.y != 0 || i == j0 + 1) ? pii + pjv.y + ev.y * vp : ninf;
  r.z = (av.z != 0 || i == j0 + 2) ? pii + pjv.z + ev.z * vp : ninf;
  r.w = (av.w != 0 || i == j0 + 3) ? pii + pjv.w + ev.w * vp : ninf;
  ((float4*)p)[q] = r;
}

// ---------------------------------------------------------------------------
extern "C" void kernel_launch(void* const* d_in, const int* in_sizes, int n_in,
                              void* d_out, int out_size, void* d_ws, size_t ws_size,
                              hipStream_t stream) {
  const float* x   = (const float*)d_in[0];
  const float* h   = (const float*)d_in[1];
  const int*   adj = (const int*)d_in[2];
  const float* e   = (const float*)d_in[3];
  const float* Wn  = (const float*)d_in[4];
  const float* We  = (const float*)d_in[5];
  const float* Wm  = (const float*)d_in[6];
  const float* Wu  = (const float*)d_in[7];
  const float* Wtm = (const float*)d_in[8];
  const float* Wtu = (const float*)d_in[9];
  const float* Wd  = (const float*)d_in[10];
  const float* Wt  = (const float*)d_in[11];
  const float* Wp  = (const float*)d_in[12];

  float* out   = (float*)d_out;
  float* new_x = out;                       // 6144
  float* p     = out + 6144;                // 1048576
  float* tau   = out + 6144 + 1048576;      // 4
  float* nh    = out + 6144 + 1048576 + 4;  // 65536

  float* ws  = (float*)d_ws;
  float* z   = ws;                 // 65536
  float* t1  = ws + 65536;         // zi_m / hi_m
  float* t2  = ws + 131072;        // zj_m / hj_m
  float* agg = ws + 196608;        // agg1 / agg2
  float* pi  = ws + 262144;        // 2048
  float* pj  = ws + 264192;        // 2048
  float* sm  = ws + 266240;        // 160 (vm|vtm|u1|u2|vp)

  dim3 mmGrid(NB * NN / 16, 2);    // 128 row tiles x 2 col tiles

  pre_kernel<<<1, 32, 0, stream>>>(We, Wm, Wtm, Wp, Wtu, Wt, sm);
  z_wmma<<<mmGrid, 32, 0, stream>>>(x, h, Wn, z);
  // message projections for MPNN #1
  mm32_wmma<<<mmGrid, 32, 0, stream>>>(z, Wm,        nullptr, nullptr, nullptr, t1);
  mm32_wmma<<<mmGrid, 32, 0, stream>>>(z, Wm + 1024, nullptr, nullptr, nullptr, t2);
  agg_kernel<<<NB * NN / 8, dim3(32, 8), 0, stream>>>(t1, t2, sm, e, adj, agg);
  // new_h = z@Wu_z + agg@Wu_a + z
  mm32_wmma<<<mmGrid, 32, 0, stream>>>(z, Wu, agg, Wu + 1024, z, nh);
  newx_kernel<<<(NB * NN * 3 + 255) / 256, 256, 0, stream>>>(z, nh, Wd, new_x);
  // termination MPNN
  mm32_wmma<<<mmGrid, 32, 0, stream>>>(nh, Wtm,        nullptr, nullptr, nullptr, t1);
  mm32_wmma<<<mmGrid, 32, 0, stream>>>(nh, Wtm + 1024, nullptr, nullptr, nullptr, t2);
  agg_kernel<<<NB * NN / 8, dim3(32, 8), 0, stream>>>(t1, t2, sm + 32, e, adj, agg);
  tau_kernel<<<NB, 256, 0, stream>>>(nh, agg, sm, tau);
  // predecessor head
  pipj_kernel<<<NB * NN / 8, dim3(32, 8), 0, stream>>>(nh, Wp, pi, pj);
  p_kernel<<<NB * NN * NN / 4 / 256, 256, 0, stream>>>(pi, pj, e, adj, sm + 128, p);
}